// Keyattention_56487409877260
// MI455X (gfx1250) — compile-verified
//
#include <hip/hip_runtime.h>

typedef __attribute__((ext_vector_type(2))) float v2f;
typedef __attribute__((ext_vector_type(4))) float v4f;
typedef __attribute__((ext_vector_type(8))) float v8f;

constexpr int B    = 8;
constexpr int C    = 256;
constexpr int HW   = 128 * 128;   // 16384
constexpr int KCH  = 128;         // key channels
constexpr int P    = 30;          // prototypes
constexpr int PPAD = 32;          // padded to 2 WMMA M-tiles
constexpr int G    = 5;           // prototypes per class
constexpr float SCALE = 5.0f / 30.0f;

// ---------------------------------------------------------------------------
// Kernel 1: q[p,b,k] = proto[p,b,:]·Wq[g,k,:] + bq[g,k]   (g = p/G)
//           A[b,p,c] = q[p,b,:]·Wk[:,c]                    (key-conv folded away)
//           s0[b,p]  = q[p,b,:]·bk
// One 128-thread block per (b,p) including pad rows p=30,31 (zeroed).
// ---------------------------------------------------------------------------
__global__ __launch_bounds__(128)
void proj_kernel(const float* __restrict__ proto, const float* __restrict__ Wk,
                 const float* __restrict__ bk, const float* __restrict__ Wq,
                 const float* __restrict__ bq, float* __restrict__ q_out,
                 float* __restrict__ A_ws, float* __restrict__ s0_ws) {
    const int p = blockIdx.x & 31;
    const int b = blockIdx.x >> 5;
    const int t = threadIdx.x;               // 0..127
    float* Arow = A_ws + ((size_t)b * PPAD + p) * C;

    if (p >= P) {                            // zero pad rows (ws is poisoned)
        Arow[t] = 0.0f;
        Arow[t + 128] = 0.0f;
        if (t == 0) s0_ws[b * PPAD + p] = 0.0f;
        return;
    }

    __shared__ float q_s[KCH];
    const int g = p / G;
    const float* pr = proto + ((size_t)p * B + b) * C;
    const float* wq = Wq + ((size_t)g * KCH + t) * C;
    float acc = bq[g * KCH + t];
    for (int c = 0; c < C; ++c) acc = fmaf(pr[c], wq[c], acc);
    q_out[((size_t)p * B + b) * KCH + t] = acc;
    q_s[t] = acc;
    __syncthreads();

    for (int cc = t; cc < C; cc += 128) {    // 2 columns per thread
        float a = 0.0f;
        for (int k = 0; k < KCH; ++k) a = fmaf(q_s[k], Wk[k * C + cc], a);
        Arow[cc] = a;
    }
    if (t == 0) {
        float s = 0.0f;
        for (int k = 0; k < KCH; ++k) s = fmaf(q_s[k], bk[k], s);
        s0_ws[b * PPAD + p] = s;
    }
}

// ---------------------------------------------------------------------------
// Kernel 2: per batch, sim[b] = A[b] (32x256) x X[b] (256x16384) via
// v_wmma_f32_16x16x4_f32; then per-pixel w = 1/sum(exp(scale*(sim-max)))
// (the max over p of the softmax), then fused out = X * w.
// Block = 256 threads = 8 waves; wave owns 16 pixels x 32 protos.
// ---------------------------------------------------------------------------
__global__ __launch_bounds__(256)
void sim_kernel(const float* __restrict__ X, const float* __restrict__ A_ws,
                const float* __restrict__ s0_ws, float* __restrict__ sim_out,
                float* __restrict__ out_weighted) {
    const int b    = blockIdx.y;
    const int tile = blockIdx.x;             // 128 pixels per block
    const int t    = threadIdx.x;
    const int wave = t >> 5;
    const int lane = t & 31;
    const int half = lane >> 4;
    const int l    = lane & 15;

    __shared__ float A_lds[PPAD * C];        // 32 KB
    __shared__ float s0_lds[PPAD];

    {   // cooperative stage of A[b] into LDS (float4)
        const v4f* src = (const v4f*)(A_ws + (size_t)b * PPAD * C);
        v4f* dst = (v4f*)A_lds;
        for (int i = t; i < PPAD * C / 4; i += 256) dst[i] = src[i];
        if (t < PPAD) s0_lds[t] = s0_ws[b * PPAD + t];
    }
    __syncthreads();

    const int n0 = tile * 128 + wave * 16;   // this wave's 16 pixels
    const float* xb = X + (size_t)b * C * HW;

    // --- GEMM: 64 K-steps of 16x16x4 fp32 WMMA, two M-tiles ---
    v8f d0 = {};                             // protos 0..15
    v8f d1 = {};                             // protos 16..31 (30,31 are pad)
    #pragma unroll 4
    for (int ks = 0; ks < C; ks += 4) {
        const int kk = ks + 2 * half;        // A: lane M=l, K = 2*half + vgpr
        v2f a0 = *(const v2f*)&A_lds[l * C + kk];
        v2f a1 = *(const v2f*)&A_lds[(l + 16) * C + kk];
        const float* xp = xb + (size_t)kk * HW + n0 + l;   // B: lane N=l
        v2f bf;
        bf.x = xp[0];
        bf.y = xp[HW];
        d0 = __builtin_amdgcn_wmma_f32_16x16x4_f32(false, a0, false, bf,
                                                   (short)0, d0, false, false);
        d1 = __builtin_amdgcn_wmma_f32_16x16x4_f32(false, a1, false, bf,
                                                   (short)0, d1, false, false);
    }

    // --- add q.bk term, store sim, track max ---
    float vals0[8], vals1[8];
    float mx = -3.4e38f;
    const size_t simbase = (size_t)b * P * HW + n0 + l;
    #pragma unroll
    for (int r = 0; r < 8; ++r) {
        const int p0 = r + 8 * half;         // 0..15
        const float v = d0[r] + s0_lds[p0];
        sim_out[simbase + (size_t)p0 * HW] = v;
        vals0[r] = v;
        mx = fmaxf(mx, v);
    }
    #pragma unroll
    for (int r = 0; r < 8; ++r) {
        const int p1 = 16 + r + 8 * half;    // 16..31
        const float v = d1[r] + s0_lds[p1];
        if (p1 < P) {
            sim_out[simbase + (size_t)p1 * HW] = v;
            mx = fmaxf(mx, v);
            vals1[r] = v;
        } else {
            vals1[r] = -3.4e38f;             // excluded from softmax
        }
    }

    // --- per-pixel softmax-max: lanes l and l+16 jointly hold p=0..29 ---
    const float mt = fmaxf(mx, __shfl_xor(mx, 16, 32));
    float se = 0.0f;
    #pragma unroll
    for (int r = 0; r < 8; ++r) se += __expf(SCALE * (vals0[r] - mt));
    #pragma unroll
    for (int r = 0; r < 8; ++r) {
        const int p1 = 16 + r + 8 * half;
        if (p1 < P) se += __expf(SCALE * (vals1[r] - mt));
    }
    se += __shfl_xor(se, 16, 32);
    const float w = 1.0f / se;               // max_p softmax_p

    // --- fused weighting: out[b,c,n] = X[b,c,n] * w  (16 pixels, all c) ---
    const float* xcol = xb + n0 + l;
    float* ocol = out_weighted + (size_t)b * C * HW + n0 + l;
    #pragma unroll 4
    for (int c = half; c < C; c += 2) {
        ocol[(size_t)c * HW] = xcol[(size_t)c * HW] * w;
    }
}

// ---------------------------------------------------------------------------
extern "C" void kernel_launch(void* const* d_in, const int* in_sizes, int n_in,
                              void* d_out, int out_size, void* d_ws, size_t ws_size,
                              hipStream_t stream) {
    const float* X     = (const float*)d_in[0];  // [8,256,128,128]
    const float* proto = (const float*)d_in[1];  // [30,8,256]
    const float* Wk    = (const float*)d_in[2];  // [128,256]
    const float* bk    = (const float*)d_in[3];  // [128]
    const float* Wq    = (const float*)d_in[4];  // [6,128,256]
    const float* bq    = (const float*)d_in[5];  // [6,128]

    float* out          = (float*)d_out;
    float* out_weighted = out;                               // 8*256*16384
    float* q_out        = out + (size_t)B * C * HW;          // 30*8*128
    float* sim_out      = q_out + (size_t)P * B * KCH;       // 8*30*16384

    float* ws    = (float*)d_ws;
    float* A_ws  = ws;                                       // 8*32*256 floats
    float* s0_ws = ws + (size_t)B * PPAD * C;                // 256 floats

    proj_kernel<<<B * PPAD, 128, 0, stream>>>(proto, Wk, bk, Wq, bq,
                                              q_out, A_ws, s0_ws);
    sim_kernel<<<dim3(HW / 128, B), 256, 0, stream>>>(X, A_ws, s0_ws,
                                                      sim_out, out_weighted);
}